// SpinConvTri2d_12369505813007
// MI455X (gfx1250) — compile-verified
//
#include <hip/hip_runtime.h>
#include <hip/hip_bf16.h>

typedef float v2f __attribute__((ext_vector_type(2)));
typedef float v8f __attribute__((ext_vector_type(8)));

#define KOFF 19
#define MULC 16
#define WST  18                  // padded per-tensor row stride (floats)
// interleaved layout: lds[ ((k*16 + w)*5 + t)*WST + u ]  -> 90 dwords per (k,w)

// Hexagonal stencil offsets (KS=2), exact order of _hex_offsets
__device__ __constant__ int2 c_off[KOFF] = {
    {0,0},
    {1,1},{1,0},{0,-1},{-1,-1},{-1,0},{0,1},
    {2,2},{2,1},{2,0},{1,-1},{0,-2},{-1,-2},{-2,-2},{-2,-1},{-2,0},{-1,1},{0,2},{1,2}
};

__global__ __launch_bounds__(256) void spinconv_tri2d_kernel(
    const float* __restrict__ feat, const float* __restrict__ spin,
    const float* __restrict__ w000, const float* __restrict__ w011,
    const float* __restrict__ w101, const float* __restrict__ w110,
    const float* __restrict__ w111, float* __restrict__ out)
{
    // ---- stage all 5 weight tensors into LDS, transposed + interleaved ----
    // tensor slot order: 0:w000 1:w110 2:w011 3:w101 4:w111
    __shared__ float lds_w[KOFF * MULC * 5 * WST];   // 109,440 bytes (< 320 KB/WGP)
    {
        const float* wsrc[5] = { w000, w110, w011, w101, w111 };
#pragma unroll
        for (int t = 0; t < 5; ++t) {
            const float* src = wsrc[t];
#pragma unroll 1
            for (int r = threadIdx.x; r < KOFF * 256; r += 256) {
                const int k = r >> 8;
                const int u = (r >> 4) & 15;
                const int w = r & 15;
                lds_w[((k * MULC + w) * 5 + t) * WST + u] = src[r];
            }
        }
    }
    __syncthreads();

    const int lane = threadIdx.x & 31;
    const int wave = threadIdx.x >> 5;
    const int tile = blockIdx.x * 8 + wave;   // one wave = 16 consecutive sites
    const int site0 = tile * 16;
    const int n  = site0 >> 14;               // /(128*128)
    const int x  = (site0 >> 7) & 127;
    const int y0 = site0 & 127;               // multiple of 16
    const int m    = lane & 15;               // A: site row; B/D: output column
    const int half = lane >> 4;
    const int y    = y0 + m;

    // sh1 = C1 * spin[[1,2,0]] for this lane's site
    const float C1 = 0.4886025119029199f;
    const float* sp = spin + (size_t)(site0 + m) * 3;
    const float sh1_0 = C1 * sp[1];
    const float sh1_1 = C1 * sp[2];
    const float sh1_2 = C1 * sp[0];

    v8f acc000 = {}; v8f acc110 = {}; v8f acc011 = {};
    v8f acc101_0 = {}; v8f acc101_1 = {}; v8f acc101_2 = {};
    v8f acc111_0 = {}; v8f acc111_1 = {}; v8f acc111_2 = {};

    const int nbase = n << 14;

#pragma unroll 1
    for (int k = 0; k < KOFF; ++k) {
        const int2 o  = c_off[k];
        const int xn = (x + o.x) & 127;
        const int yn = (y + o.y) & 127;
        const float* fp = feat + (size_t)(nbase + (xn << 7) + yn) * (4 * MULC);
        // single LDS base; all 20 fragment loads use immediate offsets (< 360 B)
        const float* wl = lds_w + ((k * MULC + m) * 5) * WST + 2 * half;
#pragma unroll
        for (int ub = 0; ub < 4; ++ub) {
            const int ua = ub * 4 + half * 2;     // lane's first K index in this slab
            // ---- A fragments: s, v, v.sh1, cross(v, sh1) for two K's ----
            const float s0  = fp[ua];
            const float s1  = fp[ua + 1];
            const float* vp = fp + MULC + 3 * ua; // 6 contiguous floats (u, u+1)
            const float va0 = vp[0], va1 = vp[1], va2 = vp[2];
            const float vb0 = vp[3], vb1 = vp[4], vb2 = vp[5];
            v2f a_s  = { s0, s1 };
            v2f a_d  = { va0*sh1_0 + va1*sh1_1 + va2*sh1_2,
                         vb0*sh1_0 + vb1*sh1_1 + vb2*sh1_2 };
            v2f a_v0 = { va0, vb0 };
            v2f a_v1 = { va1, vb1 };
            v2f a_v2 = { va2, vb2 };
            v2f a_c0 = { va1*sh1_2 - va2*sh1_1, vb1*sh1_2 - vb2*sh1_1 };
            v2f a_c1 = { va2*sh1_0 - va0*sh1_2, vb2*sh1_0 - vb0*sh1_2 };
            v2f a_c2 = { va0*sh1_1 - va1*sh1_0, vb0*sh1_1 - vb1*sh1_0 };
            // ---- B fragments: contiguous u-pairs, interleaved tensors ----
            v2f b000 = *(const v2f*)(wl + 0 * WST + 4 * ub);
            v2f b110 = *(const v2f*)(wl + 1 * WST + 4 * ub);
            v2f b011 = *(const v2f*)(wl + 2 * WST + 4 * ub);
            v2f b101 = *(const v2f*)(wl + 3 * WST + 4 * ub);
            v2f b111 = *(const v2f*)(wl + 4 * WST + 4 * ub);

            acc000 = __builtin_amdgcn_wmma_f32_16x16x4_f32(false, a_s,  false, b000, (short)0, acc000, false, false);
            acc110 = __builtin_amdgcn_wmma_f32_16x16x4_f32(false, a_d,  false, b110, (short)0, acc110, false, false);
            acc011 = __builtin_amdgcn_wmma_f32_16x16x4_f32(false, a_s,  false, b011, (short)0, acc011, false, false);
            acc101_0 = __builtin_amdgcn_wmma_f32_16x16x4_f32(false, a_v0, false, b101, (short)0, acc101_0, false, false);
            acc101_1 = __builtin_amdgcn_wmma_f32_16x16x4_f32(false, a_v1, false, b101, (short)0, acc101_1, false, false);
            acc101_2 = __builtin_amdgcn_wmma_f32_16x16x4_f32(false, a_v2, false, b101, (short)0, acc101_2, false, false);
            acc111_0 = __builtin_amdgcn_wmma_f32_16x16x4_f32(false, a_c0, false, b111, (short)0, acc111_0, false, false);
            acc111_1 = __builtin_amdgcn_wmma_f32_16x16x4_f32(false, a_c1, false, b111, (short)0, acc111_1, false, false);
            acc111_2 = __builtin_amdgcn_wmma_f32_16x16x4_f32(false, a_c2, false, b111, (short)0, acc111_2, false, false);
        }
    }

    // ---- epilogue: combine channels, write (16 sites) x (64 ch) tile ----
    const float C0        = 0.28209479177387814f;
    const float A_S       = 0.04055535739f;   // 1/sqrt(2*19*16)
    const float A_V       = 0.03311331098f;   // 1/sqrt(3*19*16)
    const float INV_SQRT3 = 0.57735026919f;
    const float INV_SQRT2 = 0.70710678119f;

    float* outp = out + (size_t)site0 * 64;
#pragma unroll
    for (int j = 0; j < 8; ++j) {
        const int M = j + half * 8;           // site row held by this lane/vgpr
        // sh1 of site M lives in lane M (0..15)
        const float s1m0 = __shfl(sh1_0, M, 32);
        const float s1m1 = __shfl(sh1_1, M, 32);
        const float s1m2 = __shfl(sh1_2, M, 32);
        float* op = outp + (size_t)M * 64;
        op[m] = A_S * (C0 * acc000[j] + INV_SQRT3 * acc110[j]);
        const float t011 = acc011[j];
        float* ov = op + MULC + 3 * m;
        ov[0] = A_V * (s1m0 * t011 + C0 * acc101_0[j] + INV_SQRT2 * acc111_0[j]);
        ov[1] = A_V * (s1m1 * t011 + C0 * acc101_1[j] + INV_SQRT2 * acc111_1[j]);
        ov[2] = A_V * (s1m2 * t011 + C0 * acc101_2[j] + INV_SQRT2 * acc111_2[j]);
    }
}

extern "C" void kernel_launch(void* const* d_in, const int* in_sizes, int n_in,
                              void* d_out, int out_size, void* d_ws, size_t ws_size,
                              hipStream_t stream) {
    (void)n_in; (void)d_ws; (void)ws_size; (void)out_size;
    const float* feat = (const float*)d_in[0];
    const float* spin = (const float*)d_in[1];
    const float* w000 = (const float*)d_in[2];
    const float* w011 = (const float*)d_in[3];
    const float* w101 = (const float*)d_in[4];
    const float* w110 = (const float*)d_in[5];
    const float* w111 = (const float*)d_in[6];
    float* out = (float*)d_out;

    const int sites  = in_sizes[1] / 3;       // N*Lx*Ly = 65536
    const int tiles  = sites / 16;            // 4096 (one wave each)
    const int blocks = tiles / 8;             // 8 waves / 256-thread block

    spinconv_tri2d_kernel<<<blocks, 256, 0, stream>>>(
        feat, spin, w000, w011, w101, w110, w111, out);
}